// GCN_2834678415609
// MI455X (gfx1250) — compile-verified
//
#include <hip/hip_runtime.h>
#include <hip/hip_bf16.h>

#define NFEAT 512
#define NHID  256
#define NCLASS 64
#define KC    32     // K-chunk staged in LDS per iteration

typedef __attribute__((ext_vector_type(2))) float v2f;
typedef __attribute__((ext_vector_type(8))) float v8f;
typedef __attribute__((ext_vector_type(4))) int   v4i;

// ---- gfx1250 async global->LDS copy (ASYNCcnt) with safe fallback --------
#if defined(__has_builtin)
#if __has_builtin(__builtin_amdgcn_global_load_async_to_lds_b128)
#define HAVE_ASYNC_LDS 1
#endif
#if __has_builtin(__builtin_amdgcn_s_wait_asynccnt)
#define HAVE_WAIT_ASYNC_BUILTIN 1
#endif
#endif

#ifdef HAVE_ASYNC_LDS
// per clang diagnostic: param 0 is `v4i __device__*` i.e. addrspace(1) v4i*
#define GCN_GPTR(p) ((__attribute__((address_space(1))) v4i*)(p))
#define GCN_LPTR(p) ((__attribute__((address_space(3))) v4i*)(p))
__device__ __forceinline__ void gcn_wait_async() {
#ifdef HAVE_WAIT_ASYNC_BUILTIN
    __builtin_amdgcn_s_wait_asynccnt(0);
#else
    asm volatile("s_wait_asynccnt 0x0" ::: "memory");
#endif
}
#endif

// ---------------------------------------------------------------------------
// zero a float buffer (float4 stores)
// ---------------------------------------------------------------------------
__global__ void gcn_zero_f4(float4* __restrict__ p, int n4) {
    int i = blockIdx.x * blockDim.x + threadIdx.x;
    if (i < n4) p[i] = make_float4(0.f, 0.f, 0.f, 0.f);
}

// ---------------------------------------------------------------------------
// fp32 WMMA GEMM: C(MxN) = A(MxK) @ B(KxN), row-major. N % 64 == 0, K % KC == 0.
// Block = 256 threads (8 waves). Block tile = 128 rows x 64 cols:
//   wave w -> rows [(blockIdx.x*8+w)*16, +16), cols [blockIdx.y*64, +64).
// B panel (KC x 64) double-buffered in LDS, staged via async global->LDS
// (GLOBAL_LOAD_ASYNC_TO_LDS_B128 + s_wait_asynccnt). A fragments for the
// whole K-chunk are prefetched into registers so their load latency is
// covered once per chunk instead of once per WMMA group.
// ---------------------------------------------------------------------------
__global__ void __launch_bounds__(256)
gcn_gemm_wmma_f32(const float* __restrict__ A, const float* __restrict__ B,
                  float* __restrict__ C, int M, int K, int Nc) {
    __shared__ float sb[2][KC * 64];

    const int lane = threadIdx.x & 31;
    const int wave = threadIdx.x >> 5;
    const int row0 = (blockIdx.x * 8 + wave) * 16;
    const int col0 = blockIdx.y * 64;

    const int m  = lane & 15;            // row in A-tile / col lane in B,C
    const int kh = (lane >> 4) * 2;      // K sub-offset (0|2) per half-wave

    int rA = row0 + m; if (rA >= M) rA = M - 1;   // clamp keeps EXEC all-1s
    const float* arow = A + (size_t)rA * K;

    v8f acc[4];
#pragma unroll
    for (int j = 0; j < 4; ++j) acc[j] = (v8f){};

    const int nk = K / KC;

    // cooperative stage of chunk `k0` into buffer `buf` (KC*64 floats,
    // 256 threads x 2 x b128 async copies)
    auto stage = [&](int buf, int k0) {
#pragma unroll
        for (int f = threadIdx.x; f < KC * 16; f += 256) {
            int r  = f >> 4;           // 0..KC-1
            int c4 = (f & 15) * 4;     // 0..60
#ifdef HAVE_ASYNC_LDS
            __builtin_amdgcn_global_load_async_to_lds_b128(
                GCN_GPTR(&B[(size_t)(k0 + r) * Nc + col0 + c4]),
                GCN_LPTR(&sb[buf][r * 64 + c4]), 0, 0);
#else
            *(float4*)&sb[buf][r * 64 + c4] =
                *(const float4*)&B[(size_t)(k0 + r) * Nc + col0 + c4];
#endif
        }
    };

    stage(0, 0);
#ifdef HAVE_ASYNC_LDS
    gcn_wait_async();
#endif
    __syncthreads();

    for (int kc = 0; kc < nk; ++kc) {
        if (kc + 1 < nk) stage((kc + 1) & 1, (kc + 1) * KC);

        const float* sbuf  = &sb[kc & 1][0];
        const int    kbase = kc * KC;

        // prefetch all A fragments for this chunk (KC/4 x b64, one clause)
        v2f af[KC / 4];
#pragma unroll
        for (int kk = 0; kk < KC; kk += 4) {
            af[kk >> 2].x = arow[kbase + kk + kh];
            af[kk >> 2].y = arow[kbase + kk + kh + 1];
        }

#pragma unroll
        for (int kk = 0; kk < KC; kk += 4) {
            v2f a = af[kk >> 2];
#pragma unroll
            for (int j = 0; j < 4; ++j) {
                v2f b; b.x = sbuf[(kk + kh)     * 64 + j * 16 + m];
                       b.y = sbuf[(kk + kh + 1) * 64 + j * 16 + m];
                acc[j] = __builtin_amdgcn_wmma_f32_16x16x4_f32(
                             false, a, false, b, (short)0, acc[j], false, false);
            }
        }
#ifdef HAVE_ASYNC_LDS
        gcn_wait_async();          // prefetched panel landed in LDS
#endif
        __syncthreads();           // all waves done reading old panel
    }

    // store 4 tiles, guarded per row (divergent stores are fine post-WMMA)
    const int rbase = row0 + (lane >> 4) * 8;
#pragma unroll
    for (int j = 0; j < 4; ++j) {
        float* cc = C + col0 + j * 16 + m;
#pragma unroll
        for (int v = 0; v < 8; ++v) {
            int r = rbase + v;
            if (r < M) cc[(size_t)r * Nc] = acc[j][v];
        }
    }
}

// ---------------------------------------------------------------------------
// SpMM via edge-parallel scatter-atomics: dst[row[e]] += val[e] * src[col[e]]
// One wave per edge; gathers hit L2 (src fits in 192MB L2), atomics at L2.
// ---------------------------------------------------------------------------
__global__ void gcn_spmm256(const int* __restrict__ rows,
                            const int* __restrict__ cols,
                            const float* __restrict__ vals,
                            const float* __restrict__ src,
                            float* __restrict__ dst, int E) {
    int e    = (int)((blockIdx.x * blockDim.x + threadIdx.x) >> 5);
    int lane = threadIdx.x & 31;
    if (e >= E) return;
    int r = rows[e], c = cols[e];
    float v = vals[e];
    const float4* s = (const float4*)(src + (size_t)c * NHID);
    float4 x0 = s[lane];
    float4 x1 = s[lane + 32];
    float* d = dst + (size_t)r * NHID + lane * 4;
    atomicAdd(d + 0, v * x0.x); atomicAdd(d + 1, v * x0.y);
    atomicAdd(d + 2, v * x0.z); atomicAdd(d + 3, v * x0.w);
    float* d2 = d + 128;
    atomicAdd(d2 + 0, v * x1.x); atomicAdd(d2 + 1, v * x1.y);
    atomicAdd(d2 + 2, v * x1.z); atomicAdd(d2 + 3, v * x1.w);
}

__global__ void gcn_spmm64(const int* __restrict__ rows,
                           const int* __restrict__ cols,
                           const float* __restrict__ vals,
                           const float* __restrict__ src,
                           float* __restrict__ dst, int E) {
    int e    = (int)((blockIdx.x * blockDim.x + threadIdx.x) >> 5);
    int lane = threadIdx.x & 31;
    if (e >= E) return;
    int r = rows[e], c = cols[e];
    float v = vals[e];
    const float2* s = (const float2*)(src + (size_t)c * NCLASS);
    float2 x = s[lane];
    float* d = dst + (size_t)r * NCLASS + lane * 2;
    atomicAdd(d + 0, v * x.x);
    atomicAdd(d + 1, v * x.y);
}

// ---------------------------------------------------------------------------
// h = relu(h + b1), 256 features, float4-vectorized in place
// ---------------------------------------------------------------------------
__global__ void gcn_bias_relu256(float* __restrict__ h,
                                 const float* __restrict__ b, int n4) {
    int i = blockIdx.x * blockDim.x + threadIdx.x;
    if (i >= n4) return;
    float4 x  = ((float4*)h)[i];
    float4 bb = ((const float4*)b)[i & 63];
    x.x = fmaxf(x.x + bb.x, 0.f);
    x.y = fmaxf(x.y + bb.y, 0.f);
    x.z = fmaxf(x.z + bb.z, 0.f);
    x.w = fmaxf(x.w + bb.w, 0.f);
    ((float4*)h)[i] = x;
}

// ---------------------------------------------------------------------------
// Fused head: logits = (s1 + b2) @ LW + Lb, then row-wise log_softmax.
// One wave computes a 16x64 block (4 WMMA tiles), then reduces each row
// inside its 16-lane half via shfl_xor (ds_swizzle/bpermute).
// ---------------------------------------------------------------------------
__global__ void gcn_final_logsoftmax(const float* __restrict__ s1,
                                     const float* __restrict__ b2,
                                     const float* __restrict__ LW,
                                     const float* __restrict__ Lb,
                                     float* __restrict__ out,
                                     int M, int Mtiles) {
    int wave = (int)((blockIdx.x * blockDim.x + threadIdx.x) >> 5);
    int lane = threadIdx.x & 31;
    if (wave >= Mtiles) return;
    int row0 = wave * 16;

    int m  = lane & 15;
    int kh = (lane >> 4) * 2;

    int rA = row0 + m; if (rA >= M) rA = M - 1;
    const float* arow = s1 + (size_t)rA * NCLASS;

    v8f acc[4];
#pragma unroll
    for (int j = 0; j < 4; ++j) acc[j] = (v8f){};

#pragma unroll
    for (int k = 0; k < NCLASS; k += 4) {
        v2f a; a.x = arow[k + kh]     + b2[k + kh];
               a.y = arow[k + kh + 1] + b2[k + kh + 1];
#pragma unroll
        for (int j = 0; j < 4; ++j) {
            v2f b; b.x = LW[(k + kh)     * NCLASS + j * 16 + m];
                   b.y = LW[(k + kh + 1) * NCLASS + j * 16 + m];
            acc[j] = __builtin_amdgcn_wmma_f32_16x16x4_f32(
                         false, a, false, b, (short)0, acc[j], false, false);
        }
    }

#pragma unroll
    for (int j = 0; j < 4; ++j) {
        float lb = Lb[j * 16 + m];
#pragma unroll
        for (int v = 0; v < 8; ++v) acc[j][v] += lb;
    }

#pragma unroll
    for (int v = 0; v < 8; ++v) {
        float mx = fmaxf(fmaxf(acc[0][v], acc[1][v]),
                         fmaxf(acc[2][v], acc[3][v]));
        for (int off = 1; off < 16; off <<= 1)
            mx = fmaxf(mx, __shfl_xor(mx, off, 32));
        float s = 0.f;
#pragma unroll
        for (int j = 0; j < 4; ++j) s += expf(acc[j][v] - mx);
        for (int off = 1; off < 16; off <<= 1)
            s += __shfl_xor(s, off, 32);
        float lse = mx + logf(s);
        int r = row0 + v + (lane >> 4) * 8;
        if (r < M) {
#pragma unroll
            for (int j = 0; j < 4; ++j)
                out[(size_t)r * NCLASS + j * 16 + m] = acc[j][v] - lse;
        }
    }
}

// ---------------------------------------------------------------------------
extern "C" void kernel_launch(void* const* d_in, const int* in_sizes, int n_in,
                              void* d_out, int out_size, void* d_ws, size_t ws_size,
                              hipStream_t stream) {
    const float* x   = (const float*)d_in[0];
    const float* a0v = (const float*)d_in[1];
    const float* a1v = (const float*)d_in[2];
    const float* W1  = (const float*)d_in[3];
    const float* b1  = (const float*)d_in[4];
    const float* W2  = (const float*)d_in[5];
    const float* b2  = (const float*)d_in[6];
    const float* LW  = (const float*)d_in[7];
    const float* Lb  = (const float*)d_in[8];
    const int* a0r   = (const int*)d_in[9];
    const int* a0c   = (const int*)d_in[10];
    const int* a1r   = (const int*)d_in[11];
    const int* a1c   = (const int*)d_in[12];
    float* out = (float*)d_out;

    const int Nn = in_sizes[0] / NFEAT;   // 50000
    const int E0 = in_sizes[1];
    const int E1 = in_sizes[2];

    // workspace (floats): [ t : Nn*NHID | s0 : Nn*NHID ]; t's region is
    // reused for u and s1 once spmm0 has consumed t.
    float* t  = (float*)d_ws;
    float* s0 = t + (size_t)Nn * NHID;
    float* u  = t;
    float* s1 = t + (size_t)Nn * NCLASS;

    const int BLK = 256;                       // 8 waves
    const int MB  = (Nn + 127) / 128;          // 128-row GEMM blocks
    const int MT  = (Nn + 15) / 16;            // 16-row tiles (head kernel)

    // 1) zero accumulator s0
    {
        int n4 = (Nn * NHID) / 4;
        gcn_zero_f4<<<(n4 + BLK - 1) / BLK, BLK, 0, stream>>>((float4*)s0, n4);
    }
    // 2) t = x @ W1   (M=Nn, K=512, N=256)
    gcn_gemm_wmma_f32<<<dim3(MB, NHID / 64), BLK, 0, stream>>>(x, W1, t,
                                                               Nn, NFEAT, NHID);
    // 3) s0 = spmm(adj0, t)
    gcn_spmm256<<<(E0 + 7) / 8, BLK, 0, stream>>>(a0r, a0c, a0v, t, s0, E0);
    // 4) s0 = relu(s0 + b1)
    {
        int n4 = (Nn * NHID) / 4;
        gcn_bias_relu256<<<(n4 + BLK - 1) / BLK, BLK, 0, stream>>>(s0, b1, n4);
    }
    // 5) u = s0 @ W2   (M=Nn, K=256, N=64)
    gcn_gemm_wmma_f32<<<dim3(MB, NCLASS / 64), BLK, 0, stream>>>(s0, W2, u,
                                                                 Nn, NHID, NCLASS);
    // 6) zero accumulator s1
    {
        int n4 = (Nn * NCLASS) / 4;
        gcn_zero_f4<<<(n4 + BLK - 1) / BLK, BLK, 0, stream>>>((float4*)s1, n4);
    }
    // 7) s1 = spmm(adj1, u)
    gcn_spmm64<<<(E1 + 7) / 8, BLK, 0, stream>>>(a1r, a1c, a1v, u, s1, E1);
    // 8) out = log_softmax((s1 + b2) @ LW + Lb)
    gcn_final_logsoftmax<<<(MT + 7) / 8, BLK, 0, stream>>>(s1, b2, LW, Lb, out,
                                                           Nn, MT);
}